// DecoderRNN_9723805958326
// MI455X (gfx1250) — compile-verified
//
#include <hip/hip_runtime.h>
#include <cmath>

#define H 2048
#define S 4096
#define V 50257

typedef __attribute__((ext_vector_type(2))) float v2f;
typedef __attribute__((ext_vector_type(8))) float v8f;

__device__ __forceinline__ float sigm(float x) { return 1.0f / (1.0f + expf(-x)); }

// ---------------------------------------------------------------------------
// WMMA f32 GEMV: out[m] = act( dot(W[m, 0:K], x) + bias[m] )
// One wave per 16-row tile, V_WMMA_F32_16X16X4_F32, N-column 0 extracted.
// Block = 256 threads = 8 waves = 128 rows. K must be a multiple of 16.
// ---------------------------------------------------------------------------
template <int ACT>
__global__ void __launch_bounds__(256)
wmma_gemv(const float* __restrict__ W, const float* __restrict__ x,
          const float* __restrict__ bias, float* __restrict__ out,
          int M, int K) {
  __shared__ float xs[4096];  // stage the input vector (K <= 4096)

  const int tx = threadIdx.x;
  for (int i = tx * 4; i < K; i += 256 * 4) {
    *(float4*)(xs + i) = *(const float4*)(x + i);
  }
  __syncthreads();

  const int wave = tx >> 5;
  const int lane = tx & 31;
  const int tile = blockIdx.x * 8 + wave;
  const int m0 = tile * 16;

  if (m0 < M) {                 // wave-uniform: EXEC stays all-ones for WMMA
    const int half = lane >> 4; // which K-pair this lane supplies
    int m = m0 + (lane & 15);
    if (m >= M) m = M - 1;      // clamp tail-row loads; writes are guarded
    const float* wrow = W + (size_t)m * K;

    v8f acc = {};
    for (int kb = 0; kb < K; kb += 16) {
      // Each lane: 2 x float4 contiguous from its row; lanes 0-15 cover
      // K = kb+0..7 of rows m0..m0+15, lanes 16-31 cover K = kb+8..15.
      const float4 w0 = *(const float4*)(wrow + kb + 8 * half);
      const float4 w1 = *(const float4*)(wrow + kb + 8 * half + 4);
      const float4 x0 = *(const float4*)(xs + kb + 8 * half);
      const float4 x1 = *(const float4*)(xs + kb + 8 * half + 4);

      v2f a, b;
      a[0] = w0.x; a[1] = w0.y; b[0] = x0.x; b[1] = x0.y;
      acc = __builtin_amdgcn_wmma_f32_16x16x4_f32(false, a, false, b, (short)0, acc, false, false);
      a[0] = w0.z; a[1] = w0.w; b[0] = x0.z; b[1] = x0.w;
      acc = __builtin_amdgcn_wmma_f32_16x16x4_f32(false, a, false, b, (short)0, acc, false, false);
      a[0] = w1.x; a[1] = w1.y; b[0] = x1.x; b[1] = x1.y;
      acc = __builtin_amdgcn_wmma_f32_16x16x4_f32(false, a, false, b, (short)0, acc, false, false);
      a[0] = w1.z; a[1] = w1.w; b[0] = x1.z; b[1] = x1.w;
      acc = __builtin_amdgcn_wmma_f32_16x16x4_f32(false, a, false, b, (short)0, acc, false, false);
    }

    // D[m][0]: lane 0 holds rows m0..m0+7 in acc[0..7]; lane 16 holds m0+8..m0+15.
    if ((lane & 15) == 0) {
      const int base = m0 + 8 * half;
#pragma unroll
      for (int i = 0; i < 8; ++i) {
        const int row = base + i;
        if (row < M) {
          float v = acc[i];
          if (bias) v += bias[row];
          if (ACT == 1) v = tanhf(v);
          out[row] = v;
        }
      }
    }
  }
}

// ---------------------------------------------------------------------------
__global__ void embed_copy(const int* __restrict__ input_id,
                           const float* __restrict__ emb,
                           float* __restrict__ x) {
  const int k = blockIdx.x * 256 + threadIdx.x;
  if (k < H) x[k] = emb[(size_t)input_id[0] * H + k];
}

__global__ void gru_gates(const float* __restrict__ gi, const float* __restrict__ gh,
                          const float* __restrict__ h_in, float* __restrict__ hnew,
                          float* __restrict__ cat, float* __restrict__ out_hidden) {
  const int j = blockIdx.x * 256 + threadIdx.x;
  if (j < H) {
    const float r = sigm(gi[j] + gh[j]);
    const float z = sigm(gi[H + j] + gh[H + j]);
    const float n = tanhf(gi[2 * H + j] + r * gh[2 * H + j]);
    const float hn = (1.0f - z) * n + z * h_in[j];
    hnew[j] = hn;
    cat[H + j] = hn;   // second half of [ctx, h_new]
    cat[j] = 0.0f;     // zero ctx accumulator for atomic adds
    out_hidden[j] = hn;
  }
}

__global__ void __launch_bounds__(1024)
softmax_s(const float* __restrict__ scores, float* __restrict__ attn_ws,
          float* __restrict__ attn_out) {
  __shared__ float red[1024];
  const int t = threadIdx.x;
  float m = -3.4e38f;
  for (int s = t; s < S; s += 1024) m = fmaxf(m, scores[s]);
  red[t] = m; __syncthreads();
  for (int o = 512; o > 0; o >>= 1) { if (t < o) red[t] = fmaxf(red[t], red[t + o]); __syncthreads(); }
  const float mx = red[0]; __syncthreads();
  float sum = 0.0f;
  for (int s = t; s < S; s += 1024) sum += expf(scores[s] - mx);
  red[t] = sum; __syncthreads();
  for (int o = 512; o > 0; o >>= 1) { if (t < o) red[t] += red[t + o]; __syncthreads(); }
  const float inv = 1.0f / red[0];
  for (int s = t; s < S; s += 1024) {
    const float a = expf(scores[s] - mx) * inv;
    attn_ws[s] = a;
    attn_out[s] = a;
  }
}

// ctx[h] = sum_s attn[s] * enc[s, h]  (coalesced over h; S split across grid.y)
__global__ void ctx_accum(const float* __restrict__ enc, const float* __restrict__ attn,
                          float* __restrict__ cat) {
  const int h = blockIdx.x * 256 + threadIdx.x;
  const int s0 = blockIdx.y * 128;
  float acc = 0.0f;
  for (int s = s0; s < s0 + 128; ++s) acc = fmaf(attn[s], enc[(size_t)s * H + h], acc);
  atomicAdd(&cat[h], acc);
}

__global__ void __launch_bounds__(1024)
lse_v(const float* __restrict__ logits, float* __restrict__ stats) {
  __shared__ float red[1024];
  const int t = threadIdx.x;
  float m = -3.4e38f;
  for (int v = t; v < V; v += 1024) m = fmaxf(m, logits[v]);
  red[t] = m; __syncthreads();
  for (int o = 512; o > 0; o >>= 1) { if (t < o) red[t] = fmaxf(red[t], red[t + o]); __syncthreads(); }
  const float mx = red[0]; __syncthreads();
  float sum = 0.0f;
  for (int v = t; v < V; v += 1024) sum += expf(logits[v] - mx);
  red[t] = sum; __syncthreads();
  for (int o = 512; o > 0; o >>= 1) { if (t < o) red[t] += red[t + o]; __syncthreads(); }
  if (t == 0) { stats[0] = mx; stats[1] = logf(red[0]); }
}

__global__ void logsoftmax_write(const float* __restrict__ logits,
                                 const float* __restrict__ stats,
                                 float* __restrict__ out) {
  const int v = blockIdx.x * 256 + threadIdx.x;
  if (v < V) out[v] = logits[v] - stats[0] - stats[1];
}

// ---------------------------------------------------------------------------
extern "C" void kernel_launch(void* const* d_in, const int* in_sizes, int n_in,
                              void* d_out, int out_size, void* d_ws, size_t ws_size,
                              hipStream_t stream) {
  const int*   input_id = (const int*)d_in[0];
  const float* hidden   = (const float*)d_in[1];
  const float* enc      = (const float*)d_in[2];
  const float* emb      = (const float*)d_in[3];
  const float* W_ih     = (const float*)d_in[4];
  const float* W_hh     = (const float*)d_in[5];
  const float* b_ih     = (const float*)d_in[6];
  const float* b_hh     = (const float*)d_in[7];
  const float* W_comb   = (const float*)d_in[8];
  const float* b_comb   = (const float*)d_in[9];
  const float* W_out    = (const float*)d_in[10];
  const float* b_out    = (const float*)d_in[11];

  float* out        = (float*)d_out;
  float* out_logp   = out;            // [V]
  float* out_hidden = out + V;        // [H]
  float* out_attn   = out + V + H;    // [S]

  float* ws     = (float*)d_ws;       // 64B-aligned offsets
  float* x_emb  = ws + 0;             // 2048
  float* gi     = ws + 2048;          // 6144
  float* gh     = ws + 8192;          // 6144
  float* hnew   = ws + 14336;         // 2048
  float* cat    = ws + 16384;         // 4096 : [ctx | h_new]
  float* scores = ws + 20480;         // 4096
  float* attn   = ws + 24576;         // 4096
  float* comb   = ws + 28672;         // 2048
  float* logits = ws + 30720;         // 50257
  float* stats  = ws + 81024;         // 2

  embed_copy<<<8, 256, 0, stream>>>(input_id, emb, x_emb);
  wmma_gemv<0><<<48, 256, 0, stream>>>(W_ih, x_emb, b_ih, gi, 3 * H, H);
  wmma_gemv<0><<<48, 256, 0, stream>>>(W_hh, hidden, b_hh, gh, 3 * H, H);
  gru_gates<<<8, 256, 0, stream>>>(gi, gh, hidden, hnew, cat, out_hidden);
  wmma_gemv<0><<<32, 256, 0, stream>>>(enc, hnew, nullptr, scores, S, H);
  softmax_s<<<1, 1024, 0, stream>>>(scores, attn, out_attn);
  ctx_accum<<<dim3(8, 32), 256, 0, stream>>>(enc, attn, cat);
  wmma_gemv<1><<<16, 256, 0, stream>>>(W_comb, cat, b_comb, comb, H, 2 * H);
  wmma_gemv<0><<<393, 256, 0, stream>>>(W_out, comb, b_out, logits, V, H);
  lse_v<<<1, 1024, 0, stream>>>(logits, stats);
  logsoftmax_write<<<197, 256, 0, stream>>>(logits, stats, out_logp);
}